// LinearInterpTrigram_20624432956048
// MI455X (gfx1250) — compile-verified
//
#include <hip/hip_runtime.h>
#include <hip/hip_bf16.h>

typedef __attribute__((ext_vector_type(2))) float v2f;
typedef __attribute__((ext_vector_type(4))) float v4f;
typedef __attribute__((ext_vector_type(8))) float v8f;

#define VSZ 512
#define LSZ 8192
#define NROW (LSZ - 1)
#define B1 128            // blocks in reduce pass
#define T1 256            // threads per block (8 waves)
#define WAVES_TOTAL (B1 * (T1 / 32))

// ws layout (floats):
//   [0 .. B1-1]            per-block bigram partial sums
//   [B1 .. 2*B1-1]         per-block trigram partial sums
//   [2*B1 .. 2*B1+VSZ-1]   base[v] = w0/V + w1*uni[v]/sum(uni)
//   [2*B1+VSZ]             a2 = w2 / sum(gathered bigram rows)
//   [2*B1+VSZ+1]           a3 = w3 / sum(gathered trigram rows)

// ---------------- Pass 1: row-sum reduction via WMMA (B = ones) ----------------
__global__ __launch_bounds__(T1) void k_reduce(const int* __restrict__ batch,
                                               const float* __restrict__ bigrams,
                                               const float* __restrict__ trigrams,
                                               float* __restrict__ ws) {
    const int lane = threadIdx.x & 31;
    const int wave_in_block = threadIdx.x >> 5;
    const int wave_id = blockIdx.x * (T1 / 32) + wave_in_block;
    // A-tile (16x4 f32) load offset: lanes 0-15 hold K=0,1; lanes 16-31 hold K=2,3
    const int lane_off = (lane & 15) * 4 + (lane >> 4) * 2;

    v8f accb = {0.f, 0.f, 0.f, 0.f, 0.f, 0.f, 0.f, 0.f};
    v8f acct = {0.f, 0.f, 0.f, 0.f, 0.f, 0.f, 0.f, 0.f};
    v2f ones;
    ones.x = 1.0f;
    ones.y = 1.0f;

    // wave-uniform loop bounds -> EXEC stays all-1s for WMMA
    for (int r = wave_id; r < NROW; r += WAVES_TOTAL) {
        const int c0 = batch[r];
        const int c1 = batch[r + 1];
        const float* brow = bigrams + (size_t)c1 * VSZ;
        const float* trow = trigrams + ((size_t)c0 * VSZ + (size_t)c1) * VSZ;
#pragma unroll
        for (int c = 0; c < VSZ / 64; ++c) {
            v2f a = *(const v2f*)(brow + c * 64 + lane_off);
            accb = __builtin_amdgcn_wmma_f32_16x16x4_f32(
                false, a, false, ones, (short)0, accb, false, false);
            v2f b = *(const v2f*)(trow + c * 64 + lane_off);
            acct = __builtin_amdgcn_wmma_f32_16x16x4_f32(
                false, b, false, ones, (short)0, acct, false, false);
        }
    }

    // Sum of all 256 accumulator elements = 16 * (true chunk sum)
    float sb = 0.f, st = 0.f;
#pragma unroll
    for (int i = 0; i < 8; ++i) {
        sb += accb[i];
        st += acct[i];
    }
    for (int off = 16; off > 0; off >>= 1) {
        sb += __shfl_xor(sb, off, 32);
        st += __shfl_xor(st, off, 32);
    }

    __shared__ float lsb[T1 / 32];
    __shared__ float lst[T1 / 32];
    if (lane == 0) {
        lsb[wave_in_block] = sb * (1.0f / 16.0f);
        lst[wave_in_block] = st * (1.0f / 16.0f);
    }
    __syncthreads();
    if (threadIdx.x == 0) {
        float tb = 0.f, tt = 0.f;
        for (int i = 0; i < T1 / 32; ++i) {
            tb += lsb[i];
            tt += lst[i];
        }
        ws[blockIdx.x] = tb;
        ws[B1 + blockIdx.x] = tt;
    }
}

// ---------------- Pass 2: deterministic finalize (single block) ----------------
__global__ __launch_bounds__(VSZ) void k_finalize(const float* __restrict__ unigrams,
                                                  const float* __restrict__ w,
                                                  float* __restrict__ ws) {
    __shared__ float red[VSZ];
    const int t = threadIdx.x;

    // sum(unigrams)
    red[t] = unigrams[t];
    __syncthreads();
    for (int s = VSZ / 2; s > 0; s >>= 1) {
        if (t < s) red[t] += red[t + s];
        __syncthreads();
    }
    const float s_uni = red[0];
    __syncthreads();

    // sum of bigram partials
    red[t] = (t < B1) ? ws[t] : 0.0f;
    __syncthreads();
    for (int s = VSZ / 2; s > 0; s >>= 1) {
        if (t < s) red[t] += red[t + s];
        __syncthreads();
    }
    const float s_bi = red[0];
    __syncthreads();

    // sum of trigram partials
    red[t] = (t < B1) ? ws[B1 + t] : 0.0f;
    __syncthreads();
    for (int s = VSZ / 2; s > 0; s >>= 1) {
        if (t < s) red[t] += red[t + s];
        __syncthreads();
    }
    const float s_tri = red[0];
    __syncthreads();

    const float w0 = w[0], w1 = w[1];
    ws[2 * B1 + t] = w0 * (1.0f / (float)VSZ) + w1 * unigrams[t] / s_uni;
    if (t == 0) {
        ws[2 * B1 + VSZ] = w[2] / s_bi;
        ws[2 * B1 + VSZ + 1] = w[3] / s_tri;
    }
}

// ---------------- Pass 3: bandwidth-optimal emit (b128 loads/stores) ----------------
__global__ __launch_bounds__(128) void k_out(const int* __restrict__ batch,
                                             const float* __restrict__ bigrams,
                                             const float* __restrict__ trigrams,
                                             const float* __restrict__ ws,
                                             float* __restrict__ out) {
    const int n = blockIdx.x;
    const int c0 = batch[n];
    const int c1 = batch[n + 1];
    const int v = threadIdx.x * 4;

    const float a2 = ws[2 * B1 + VSZ];
    const float a3 = ws[2 * B1 + VSZ + 1];

    v4f bs = *(const v4f*)(ws + 2 * B1 + v);
    v4f bb = *(const v4f*)(bigrams + (size_t)c1 * VSZ + v);
    v4f tb = *(const v4f*)(trigrams + ((size_t)c0 * VSZ + (size_t)c1) * VSZ + v);

    v4f o;
#pragma unroll
    for (int i = 0; i < 4; ++i)
        o[i] = __builtin_fmaf(a3, tb[i], __builtin_fmaf(a2, bb[i], bs[i]));

    *(v4f*)(out + (size_t)n * VSZ + v) = o;
}

extern "C" void kernel_launch(void* const* d_in, const int* in_sizes, int n_in,
                              void* d_out, int out_size, void* d_ws, size_t ws_size,
                              hipStream_t stream) {
    (void)in_sizes; (void)n_in; (void)out_size; (void)ws_size;
    const int*   batch    = (const int*)d_in[0];
    const float* unigrams = (const float*)d_in[1];
    const float* bigrams  = (const float*)d_in[2];
    const float* trigrams = (const float*)d_in[3];
    const float* w        = (const float*)d_in[4];
    float* out = (float*)d_out;
    float* ws  = (float*)d_ws;

    k_reduce  <<<B1,   T1,  0, stream>>>(batch, bigrams, trigrams, ws);
    k_finalize<<<1,    VSZ, 0, stream>>>(unigrams, w, ws);
    k_out     <<<NROW, 128, 0, stream>>>(batch, bigrams, trigrams, ws, out);
}